// AttentionBlock_65309272703231
// MI455X (gfx1250) — compile-verified
//
#include <hip/hip_runtime.h>
#include <math.h>

// ---------------- problem constants (match reference) ----------------
#define HIDDEN    2880
#define HEAD_DIM  64
#define N_HEADS   64
#define N_KV      8
#define Q_MULT    8
#define LEFT_CTX  128
#define WINDOW    129           // LEFT_CTX + 1
#define Q_DIM     4096          // N_HEADS * HEAD_DIM
#define KV_DIM    512           // N_KV * HEAD_DIM
#define QKV_DIM   5120          // Q_DIM + 2*KV_DIM
#define T_SEQ     2048
#define RMS_EPS   1e-5f
#define SM_SCALE  0.125f        // 1/sqrt(64)

// Use gfx1250 GLOBAL_LOAD_ASYNC_TO_LDS_B128 (ASYNCcnt) for GEMM tile staging.
// Set to 0 to fall back to the round-1 synchronous float4+ds_store path.
#define USE_ASYNC_COPY 1

typedef __attribute__((ext_vector_type(2))) float v2f;
typedef __attribute__((ext_vector_type(8))) float v8f;

// 32-bit LDS offset of a generic pointer to a __shared__ object
// (addrspace(3) pointers are byte offsets from the workgroup LDS base).
__device__ __forceinline__ unsigned lds_off(const void* p) {
  return (unsigned)(unsigned long long)(__attribute__((address_space(3))) char*)(void*)p;
}

// async 16-byte global -> LDS copy (per lane), tracked by ASYNCcnt
__device__ __forceinline__ void async_copy_b128(unsigned lds_byte_off, const float* gaddr) {
  asm volatile("global_load_async_to_lds_b128 %0, %1, off"
               :: "v"(lds_byte_off), "v"(gaddr)
               : "memory");
}
__device__ __forceinline__ void wait_async0() {
  asm volatile("s_wait_asynccnt 0" ::: "memory");
}

// ---------------- kernel 1: RMSNorm --------------------------------
__global__ __launch_bounds__(256) void rmsnorm_kernel(
    const float* __restrict__ x, const float* __restrict__ scale,
    float* __restrict__ h) {
  const int t = blockIdx.x;
  const float* row = x + (size_t)t * HIDDEN;
  float ss = 0.f;
  for (int i = threadIdx.x; i < HIDDEN; i += 256) { float v = row[i]; ss += v * v; }
  for (int off = 16; off; off >>= 1) ss += __shfl_xor(ss, off, 32);
  __shared__ float partials[8];
  const int wave = threadIdx.x >> 5, lane = threadIdx.x & 31;
  if (lane == 0) partials[wave] = ss;
  __syncthreads();
  __shared__ float rinv;
  if (wave == 0) {
    float tot = (lane < 8) ? partials[lane] : 0.f;
    for (int off = 4; off; off >>= 1) tot += __shfl_xor(tot, off, 32);
    if (lane == 0) rinv = rsqrtf(tot * (1.0f / HIDDEN) + RMS_EPS);
  }
  __syncthreads();
  const float r = rinv;
  float* hrow = h + (size_t)t * HIDDEN;
  for (int i = threadIdx.x; i < HIDDEN; i += 256) hrow[i] = row[i] * r * scale[i];
}

// ---------------- kernel 2/5: fp32 WMMA GEMM ------------------------
// C[M,N] = A[M,K] * W[N,K]^T + bias[N] (+ resid[M,N])
// Block tile 128x64, K-step 32, 8 waves (wave32): wave grid 2x4,
// each wave computes 64x16 = 4 stacked 16x16 WMMA tiles (K inner = 4).
// Both LDS tiles are stored k-contiguous (straight copies of global rows)
// so GLOBAL_LOAD_ASYNC_TO_LDS_B128 can stage them; LDS row pitch = 36
// floats (144 B) keeps b128 stores 16B-aligned and fragment reads
// conflict-free across 64 banks. Double-buffered for load/WMMA overlap.
#define BM 128
#define BN 64
#define BK 32
#define LP 36   // LDS row pitch in floats

__global__ __launch_bounds__(256) void wmma_gemm_kernel(
    const float* __restrict__ A, const float* __restrict__ W,
    const float* __restrict__ bias, const float* __restrict__ resid,
    float* __restrict__ C, int M, int N, int K) {
  __shared__ float sA[2][BM][LP];   // sA[b][m][k]
  __shared__ float sB[2][BN][LP];   // sB[b][n][k]  (k-contiguous, no transpose)

  const int tid  = threadIdx.x;
  const int wave = tid >> 5;
  const int lane = tid & 31;
  const int wr = wave >> 2;        // 0..1 : 64-row slab
  const int wc = wave & 3;         // 0..3 : 16-col slab
  const int m0 = blockIdx.y * BM;
  const int n0 = blockIdx.x * BN;
  const int lhalf = lane >> 4;     // 0/1
  const int llow  = lane & 15;

  v8f acc[4];
  acc[0] = {}; acc[1] = {}; acc[2] = {}; acc[3] = {};

  const int rowA = tid >> 3;            // 0..31
  const int c4   = (tid & 7) * 4;       // 0..28 step 4

#if USE_ASYNC_COPY
  // issue one tile's worth of async copies into buffer `buf`
  auto issue_tile = [&](int buf, int kb) {
    // A tile: 128 rows x 32 floats = 1024 b128 chunks, 4 per thread
#pragma unroll
    for (int p = 0; p < 4; p++) {
      const int r = rowA + p * 32;
      async_copy_b128(lds_off(&sA[buf][r][c4]),
                      A + (size_t)(m0 + r) * K + kb + c4);
    }
    // B tile: 64 rows x 32 floats = 512 b128 chunks, 2 per thread
#pragma unroll
    for (int p = 0; p < 2; p++) {
      const int n = rowA + p * 32;
      async_copy_b128(lds_off(&sB[buf][n][c4]),
                      W + (size_t)(n0 + n) * K + kb + c4);
    }
  };
#else
  auto issue_tile = [&](int buf, int kb) {
#pragma unroll
    for (int p = 0; p < 4; p++) {
      const int r = rowA + p * 32;
      const float4 va = *(const float4*)(A + (size_t)(m0 + r) * K + kb + c4);
      *(float4*)&sA[buf][r][c4] = va;
    }
#pragma unroll
    for (int p = 0; p < 2; p++) {
      const int n = rowA + p * 32;
      const float4 vb = *(const float4*)(W + (size_t)(n0 + n) * K + kb + c4);
      *(float4*)&sB[buf][n][c4] = vb;
    }
  };
#endif

  const int nk = K / BK;
  issue_tile(0, 0);

  for (int kt = 0; kt < nk; kt++) {
    const int buf = kt & 1;
#if USE_ASYNC_COPY
    wait_async0();                 // this wave's copies for tile `kt` landed
#endif
    __syncthreads();               // all waves' copies visible
    if (kt + 1 < nk) issue_tile(buf ^ 1, (kt + 1) * BK);

#pragma unroll
    for (int kk = 0; kk < BK; kk += 4) {
      // B fragment (4x16): VGPR0 rows {kk,kk+1} across lane halves, VGPR1 {kk+2,kk+3}
      v2f b;
      const int bn = wc * 16 + llow;
      b.x = sB[buf][bn][kk + lhalf];
      b.y = sB[buf][bn][kk + 2 + lhalf];
#pragma unroll
      for (int mt = 0; mt < 4; mt++) {
        // A fragment (16x4): lane<16 -> K={kk,kk+1}, lane>=16 -> K={kk+2,kk+3}
        v2f a;
        const int r = wr * 64 + mt * 16 + llow;
        a.x = sA[buf][r][kk + 2 * lhalf];
        a.y = sA[buf][r][kk + 2 * lhalf + 1];
        acc[mt] = __builtin_amdgcn_wmma_f32_16x16x4_f32(
            false, a, false, b, (short)0, acc[mt], false, false);
      }
    }
    __syncthreads();               // done reading `buf` before it is re-filled
  }

  // Epilogue: C/D layout — VGPR r: lanes 0-15 M=r, lanes 16-31 M=r+8; N = lane&15
  const int gn = n0 + wc * 16 + llow;
  const float bnv = bias[gn];
#pragma unroll
  for (int mt = 0; mt < 4; mt++) {
#pragma unroll
    for (int r = 0; r < 8; r++) {
      const int gm = m0 + wr * 64 + mt * 16 + r + 8 * lhalf;
      float v = acc[mt][r] + bnv;
      if (resid) v += resid[(size_t)gm * N + gn];
      C[(size_t)gm * N + gn] = v;
    }
  }
}

// ---------------- kernel 3: YaRN RoPE (in place on q,k) -------------
__global__ __launch_bounds__(256) void rope_kernel(float* __restrict__ qkv) {
  const int t = blockIdx.x;
  const float logb   = logf(150000.0f);
  const float conc   = 0.1f * logf(32.0f) + 1.0f;
  const float low    = 32.0f * logf(4096.0f / (32.0f * 6.2831853071795864f)) / logb;
  const float high   = 32.0f * logf(4096.0f / (6.2831853071795864f)) / logb;

  for (int s = threadIdx.x; s < (N_HEADS + N_KV) * 32; s += 256) {
    const int d  = s & 31;
    const int hs = s >> 5;
    const size_t base = (size_t)t * QKV_DIM +
        (hs < N_HEADS ? (size_t)hs * HEAD_DIM
                      : (size_t)Q_DIM + (size_t)(hs - N_HEADS) * HEAD_DIM);
    const float freq   = __powf(150000.0f, (float)d * (1.0f / 32.0f));
    const float interp = 1.0f / (32.0f * freq);
    const float extrap = 1.0f / freq;
    const float ramp   = ((float)d - low) / (high - low);
    const float mask   = 1.0f - fminf(fmaxf(ramp, 0.0f), 1.0f);
    const float invf   = interp * (1.0f - mask) + extrap * mask;
    const float ang    = (float)t * invf;
    const float c = cosf(ang) * conc, sn = sinf(ang) * conc;
    const float x1 = qkv[base + 2 * d], x2 = qkv[base + 2 * d + 1];
    qkv[base + 2 * d]     = x1 * c - x2 * sn;
    qkv[base + 2 * d + 1] = x2 * c + x1 * sn;
  }
}

// ---------------- kernel 4: sliding-window attention ----------------
// One wave per (token, head). All waves in a block share the same token,
// so __syncthreads() is uniform.
__global__ __launch_bounds__(256) void attn_kernel(
    const float* __restrict__ qkv, const float* __restrict__ sinks,
    float* __restrict__ attn) {
  const int t    = blockIdx.x;
  const int wave = threadIdx.x >> 5;
  const int lane = threadIdx.x & 31;
  const int h    = blockIdx.y * 8 + wave;
  const int kvh  = h >> 3;             // Q_MULT = 8
  __shared__ float sw[8][WINDOW];

  const float* qrow = qkv + (size_t)t * QKV_DIM + h * HEAD_DIM;
  const float qx = qrow[2 * lane], qy = qrow[2 * lane + 1];

  // scores
  for (int j = 0; j < WINDOW; j++) {
    const int p = t - LEFT_CTX + j;
    float s = -1e30f;
    if (p >= 0) {
      const float* kr = qkv + (size_t)p * QKV_DIM + Q_DIM + kvh * HEAD_DIM;
      float part = qx * kr[2 * lane] + qy * kr[2 * lane + 1];
      for (int off = 16; off; off >>= 1) part += __shfl_xor(part, off, 32);
      s = part * SM_SCALE;
    }
    if (lane == 0) sw[wave][j] = s;
  }
  __syncthreads();

  // softmax with sink logit (sinks[h] * ln2)
  const float snk = sinks[h] * 0.69314718055994531f;
  float m = snk;
  for (int j = lane; j < WINDOW; j += 32) m = fmaxf(m, sw[wave][j]);
  for (int off = 16; off; off >>= 1) m = fmaxf(m, __shfl_xor(m, off, 32));
  float sum = 0.f;
  for (int j = lane; j < WINDOW; j += 32) sum += __expf(sw[wave][j] - m);
  for (int off = 16; off; off >>= 1) sum += __shfl_xor(sum, off, 32);
  sum += __expf(snk - m);
  const float rden = 1.0f / sum;
  for (int j = lane; j < WINDOW; j += 32) sw[wave][j] = __expf(sw[wave][j] - m) * rden;
  __syncthreads();

  // weighted sum over V (lane owns dims 2*lane, 2*lane+1)
  float ax = 0.f, ay = 0.f;
  for (int j = 0; j < WINDOW; j++) {
    const int p = t - LEFT_CTX + j;
    if (p < 0) continue;
    const float w = sw[wave][j];
    const float* vr = qkv + (size_t)p * QKV_DIM + Q_DIM + KV_DIM + kvh * HEAD_DIM;
    ax += w * vr[2 * lane];
    ay += w * vr[2 * lane + 1];
  }
  attn[(size_t)t * Q_DIM + h * HEAD_DIM + 2 * lane]     = ax;
  attn[(size_t)t * Q_DIM + h * HEAD_DIM + 2 * lane + 1] = ay;
}

// ---------------- launcher ------------------------------------------
extern "C" void kernel_launch(void* const* d_in, const int* in_sizes, int n_in,
                              void* d_out, int out_size, void* d_ws, size_t ws_size,
                              hipStream_t stream) {
  (void)in_sizes; (void)n_in; (void)out_size; (void)ws_size;
  const float* x          = (const float*)d_in[0];
  const float* norm_scale = (const float*)d_in[1];
  const float* w_qkv      = (const float*)d_in[2];
  const float* b_qkv      = (const float*)d_in[3];
  const float* sinks      = (const float*)d_in[4];
  const float* w_out      = (const float*)d_in[5];
  const float* b_out      = (const float*)d_in[6];
  float* out = (float*)d_out;

  // workspace: h [T,HIDDEN] | qkv [T,QKV_DIM] | attn [T,Q_DIM]  (~99 MB fp32)
  float* h    = (float*)d_ws;
  float* qkv  = h + (size_t)T_SEQ * HIDDEN;
  float* attn = qkv + (size_t)T_SEQ * QKV_DIM;

  rmsnorm_kernel<<<T_SEQ, 256, 0, stream>>>(x, norm_scale, h);
  wmma_gemm_kernel<<<dim3(QKV_DIM / BN, T_SEQ / BM), 256, 0, stream>>>(
      h, w_qkv, b_qkv, nullptr, qkv, T_SEQ, QKV_DIM, HIDDEN);
  rope_kernel<<<T_SEQ, 256, 0, stream>>>(qkv);
  attn_kernel<<<dim3(T_SEQ, N_HEADS / 8), 256, 0, stream>>>(qkv, sinks, attn);
  wmma_gemm_kernel<<<dim3(HIDDEN / BN, T_SEQ / BM), 256, 0, stream>>>(
      attn, w_out, b_out, x, out, T_SEQ, HIDDEN, Q_DIM);
}